// VideoHungarianMatcher_22479858827268
// MI455X (gfx1250) — compile-verified
//
#include <hip/hip_runtime.h>
#include <math.h>

// Problem constants (from reference)
#define BATCH      64
#define NFRAMES    36
#define QN         100   // queries per frame
#define TN         20    // targets per frame
#define NCLS       92
#define NBF        (BATCH * NFRAMES)          // 2304 independent problems
#define COST_ELEMS ((size_t)NBF * QN * TN)    // 4,608,000
#define IDX_ELEMS  ((size_t)NBF * TN)         // 46,080

#define QPAD       112   // 7 WMMA M-tiles x 16 rows (rows 100..111 zeroed)
#define PROB_LD    94    // even stride: 8B-aligned (K,K+1) pairs -> ds_load_b64

typedef float v2f __attribute__((ext_vector_type(2)));
typedef float v8f __attribute__((ext_vector_type(8)));

// ---------------------------------------------------------------------------
// Kernel 1: cost tensor.  One 256-thread block (8 wave32s) per (b,f).
//   - cooperative softmax of the 100x92 logit tile into LDS (fp32),
//     zero-padded to 112 rows so WMMA A-loads need no predication
//   - cost_class via V_WMMA_F32_16X16X4_F32 (23 chained K-steps, unrolled):
//     A = prob (LDS, ds_load_b64), B = one-hot synthesized in registers
//   - L1 + GIoU fused at the accumulator write-out
// ---------------------------------------------------------------------------
__global__ __launch_bounds__(256)
void vhm_cost_kernel(const float* __restrict__ logits,   // (B, 3600, 92)
                     const float* __restrict__ pboxes,   // (B, 3600, 4)
                     const int*   __restrict__ tlabels,  // (F, B, 20)
                     const float* __restrict__ tboxes,   // (F, B, 20, 4)
                     float* __restrict__ cost_out)       // (B, F, 100, 20)
{
    const int bf   = blockIdx.x;            // 0..2303
    const int b    = bf / NFRAMES;
    const int f    = bf - b * NFRAMES;
    const int tid  = threadIdx.x;
    const int lane = tid & 31;
    const int wave = tid >> 5;              // 8 waves

    __shared__ float s_prob[QPAD * PROB_LD];  // 42.1 KB probabilities (padded)
    __shared__ float s_pb[QN * 4];            // pred boxes (cxcywh)
    __shared__ float s_tb[TN * 4];            // tgt boxes (cxcywh)
    __shared__ int   s_lab[TN];               // tgt labels

    const size_t rowbase = (size_t)b * (NFRAMES * QN) + (size_t)f * QN;
    const size_t tbase   = ((size_t)f * BATCH + b) * TN;

    // stage small operands + zero the padding rows (100..111)
    for (int i = tid; i < QN * 4; i += 256) s_pb[i]  = pboxes[rowbase * 4 + i];
    for (int i = tid; i < TN * 4; i += 256) s_tb[i]  = tboxes[tbase * 4 + i];
    for (int i = tid; i < TN;     i += 256) s_lab[i] = tlabels[tbase + i];
    for (int i = tid; i < (QPAD - QN) * PROB_LD; i += 256)
        s_prob[QN * PROB_LD + i] = 0.f;

    // softmax: one wave per row, lanes across the 92 classes (3 strided passes)
    for (int r = wave; r < QN; r += 8) {
        const float* row = logits + (rowbase + r) * NCLS;
        __builtin_prefetch(row + 8 * NCLS, 0, 1);   // global_prefetch_b8
        const float x0 = (lane      < NCLS) ? row[lane]      : -__builtin_inff();
        const float x1 = (lane + 32 < NCLS) ? row[lane + 32] : -__builtin_inff();
        const float x2 = (lane + 64 < NCLS) ? row[lane + 64] : -__builtin_inff();
        float m = fmaxf(x0, fmaxf(x1, x2));
        #pragma unroll
        for (int off = 16; off; off >>= 1) m = fmaxf(m, __shfl_xor(m, off, 32));
        const float e0 = (lane      < NCLS) ? __expf(x0 - m) : 0.f;
        const float e1 = (lane + 32 < NCLS) ? __expf(x1 - m) : 0.f;
        const float e2 = (lane + 64 < NCLS) ? __expf(x2 - m) : 0.f;
        float s = e0 + e1 + e2;
        #pragma unroll
        for (int off = 16; off; off >>= 1) s += __shfl_xor(s, off, 32);
        const float inv = 1.f / s;
        if (lane      < NCLS) s_prob[r * PROB_LD + lane]      = e0 * inv;
        if (lane + 32 < NCLS) s_prob[r * PROB_LD + lane + 32] = e1 * inv;
        if (lane + 64 < NCLS) s_prob[r * PROB_LD + lane + 64] = e2 * inv;
    }
    __syncthreads();

    // WMMA tiles: 7 M-tiles (Q padded to 112) x 2 N-tiles (T=20 -> 32) = 14
    const int hi = lane >> 4;           // 0 or 1 (lane half)
    const int ml = lane & 15;
    for (int tile = wave; tile < 14; tile += 8) {      // wave-uniform: EXEC all 1s
        const int mt = tile >> 1, nt = tile & 1;
        const int t    = nt * 16 + ml;                 // this lane's column
        const int labT = (t < TN) ? s_lab[t] : -1;
        const int m    = mt * 16 + ml;                 // this lane's A row (<112)
        const float* arow = &s_prob[m * PROB_LD];

        v8f acc = {};
        #pragma unroll
        for (int k0 = 0; k0 < NCLS; k0 += 4) {         // 23 K-steps, unrolled
            const int kb = k0 + 2 * hi;                // 8B-aligned pair
            const v2f a = *(const v2f*)(arow + kb);    // ds_load_b64
            v2f bb;
            bb.x = (labT == kb)     ? 1.f : 0.f;       // one-hot column, exact
            bb.y = (labT == kb + 1) ? 1.f : 0.f;
            acc = __builtin_amdgcn_wmma_f32_16x16x4_f32(
                      false, a, false, bb, (short)0, acc, false, false);
        }

        if (t < TN) {
            const float tcx = s_tb[t*4+0], tcy = s_tb[t*4+1];
            const float tw  = s_tb[t*4+2], th  = s_tb[t*4+3];
            const float tx0 = tcx - 0.5f*tw, ty0 = tcy - 0.5f*th;
            const float tx1 = tcx + 0.5f*tw, ty1 = tcy + 0.5f*th;
            const float tarea = (tx1 - tx0) * (ty1 - ty0);
            #pragma unroll
            for (int r = 0; r < 8; ++r) {
                const int q = mt * 16 + r + hi * 8;    // C/D layout row
                if (q < QN) {
                    const float pcx = s_pb[q*4+0], pcy = s_pb[q*4+1];
                    const float pw  = s_pb[q*4+2], ph  = s_pb[q*4+3];
                    const float l1  = fabsf(pcx-tcx) + fabsf(pcy-tcy)
                                    + fabsf(pw -tw ) + fabsf(ph -th );
                    const float px0 = pcx - 0.5f*pw, py0 = pcy - 0.5f*ph;
                    const float px1 = pcx + 0.5f*pw, py1 = pcy + 0.5f*ph;
                    const float parea = (px1 - px0) * (py1 - py0);
                    const float iw = fmaxf(fminf(px1,tx1) - fmaxf(px0,tx0), 0.f);
                    const float ih = fmaxf(fminf(py1,ty1) - fmaxf(py0,ty0), 0.f);
                    const float inter = iw * ih;
                    const float uni   = parea + tarea - inter;
                    const float iou   = inter / uni;
                    const float ew = fmaxf(fmaxf(px1,tx1) - fminf(px0,tx0), 0.f);
                    const float eh = fmaxf(fmaxf(py1,ty1) - fminf(py0,ty0), 0.f);
                    const float earea = ew * eh;
                    const float giou  = iou - (earea - uni) / earea;
                    cost_out[((size_t)bf * QN + q) * TN + t] =
                        1.0f * (-acc[r]) + 5.0f * l1 - 2.0f * giou;
                }
            }
        }
    }
}

// ---------------------------------------------------------------------------
// Kernel 2: auction assignment + stable argsort.  One wave32 per (b,f).
// Faithful port of the JAX while-loop (max bid per object, min target on
// ties, evict previous owner, eps = spread/1000, 20000-iteration cap).
// ---------------------------------------------------------------------------
__global__ __launch_bounds__(32)
void vhm_auction_kernel(const float* __restrict__ cost,   // (NBF, 100, 20)
                        int* __restrict__ pred_idx,       // (NBF, 20)
                        int* __restrict__ tgt_idx)        // (NBF, 20)
{
    const int bf   = blockIdx.x;
    const int lane = threadIdx.x;

    __shared__ float s_ben[TN * QN];      // benefit = -cost^T  (8 KB)
    __shared__ float s_price[QN];
    __shared__ float s_bestbid[QN];
    __shared__ int   s_owner[QN];
    __shared__ int   s_winner[QN];
    __shared__ int   s_objof[TN];
    __shared__ float s_bids[TN];
    __shared__ int   s_bobj[TN];

    const float* c = cost + (size_t)bf * QN * TN;
    float mx = -1e30f, mn = 1e30f;
    for (int i = lane; i < TN * QN; i += 32) {
        const int t = i / QN, q = i - t * QN;
        const float v = -c[q * TN + t];
        s_ben[i] = v;
        mx = fmaxf(mx, v); mn = fminf(mn, v);
    }
    #pragma unroll
    for (int off = 16; off; off >>= 1) {
        mx = fmaxf(mx, __shfl_xor(mx, off, 32));
        mn = fminf(mn, __shfl_xor(mn, off, 32));
    }
    const float eps = (mx - mn + 1e-6f) * 0.001f;

    for (int i = lane; i < QN; i += 32) {
        s_price[i] = 0.f; s_owner[i] = -1; s_bestbid[i] = -1e30f; s_winner[i] = TN;
    }
    if (lane < TN) s_objof[lane] = -1;
    __builtin_amdgcn_wave_barrier();

    for (int it = 0; it < 20000; ++it) {
        const bool un = (lane < TN) && (s_objof[lane] < 0);
        if (!__any(un)) break;

        // per-target top-2 of (benefit - price) over the 100 objects
        float bid = -1e30f; int bq = 0;
        if (un) {
            float v1 = -1e30f, v2 = -1e30f; int i1 = 0;
            const float* row = &s_ben[lane * QN];
            for (int q = 0; q < QN; ++q) {
                const float v = row[q] - s_price[q];
                if (v > v1)      { v2 = v1; v1 = v; i1 = q; }
                else if (v > v2) { v2 = v; }
            }
            bid = s_price[i1] + (v1 - v2) + eps;
            bq  = i1;
        }
        if (lane < TN) { s_bids[lane] = un ? bid : -1e30f; s_bobj[lane] = bq; }
        __builtin_amdgcn_wave_barrier();

        // serial conflict resolution (<= 20 bids), exact JAX semantics
        if (lane == 0) {
            for (int t = 0; t < TN; ++t) if (s_bids[t] > -1e29f) {
                const int q = s_bobj[t];
                if (s_bids[t] > s_bestbid[q]) s_bestbid[q] = s_bids[t];
            }
            for (int t = 0; t < TN; ++t) if (s_bids[t] > -1e29f) {
                const int q = s_bobj[t];
                if (s_bids[t] >= s_bestbid[q] && t < s_winner[q]) s_winner[q] = t;
            }
            for (int t = 0; t < TN; ++t) if (s_bids[t] > -1e29f) {
                const int q = s_bobj[t];
                if (s_winner[q] < TN) {           // first touch of this object
                    const int w    = s_winner[q];
                    const int prev = s_owner[q];
                    if (prev >= 0) s_objof[prev] = -1;   // evict
                    s_objof[w]   = q;
                    s_price[q]   = s_bestbid[q];
                    s_owner[q]   = w;
                    s_winner[q]  = TN;            // reset for next iteration
                    s_bestbid[q] = -1e30f;
                }
            }
        }
        __builtin_amdgcn_wave_barrier();
    }

    // stable argsort of obj_of (T=20) via O(T^2) rank; emit pred/tgt indices
    if (lane < TN) {
        const int val = s_objof[lane];
        int rank = 0;
        for (int t2 = 0; t2 < TN; ++t2) {
            const int v2 = s_objof[t2];
            rank += ((v2 < val) || (v2 == val && t2 < lane)) ? 1 : 0;
        }
        pred_idx[(size_t)bf * TN + rank] = val;
        tgt_idx [(size_t)bf * TN + rank] = lane;
    }
}

// ---------------------------------------------------------------------------
extern "C" void kernel_launch(void* const* d_in, const int* in_sizes, int n_in,
                              void* d_out, int out_size, void* d_ws, size_t ws_size,
                              hipStream_t stream)
{
    const float* logits  = (const float*)d_in[0];   // (64, 3600, 92)
    const float* pboxes  = (const float*)d_in[1];   // (64, 3600, 4)
    const int*   tlabels = (const int*)  d_in[2];   // (36, 64, 20)
    const float* tboxes  = (const float*)d_in[3];   // (36, 64, 20, 4)

    float* cost = (float*)d_out;                            // 4,608,000 f32
    int*   pred = (int*)d_out + COST_ELEMS;                 // 46,080 i32
    int*   tgt  = pred + IDX_ELEMS;                         // 46,080 i32

    vhm_cost_kernel<<<NBF, 256, 0, stream>>>(logits, pboxes, tlabels, tboxes, cost);
    vhm_auction_kernel<<<NBF, 32, 0, stream>>>(cost, pred, tgt);
}